// NNHMM_88888643158752
// MI455X (gfx1250) — compile-verified
//
#include <hip/hip_runtime.h>
#include <cstdint>
#include <cstddef>

// ---------------------------------------------------------------------------
// HMM forward pass, T=2^21 steps, K=8 states, D=4 dims.
// Linear-space chunked scan with exact power-of-two rescaling:
//   alpha^T_t = alpha^T_{t-1} @ (Tlin * diag(em_t))
// Phase 1: 65536 chunks of 32 steps, one chunk per lane (2048 waves), VALU
//          8x8 register matmuls; x_seq staged through LDS with
//          global_load_async_to_lds_b128 (double-buffered, s_wait_asynccnt).
// Phase 2: ordered tree reduction of 8x8 transfer matrices using
//          v_wmma_f32_16x16x4_f32 (Q = P^T carried in C-layout; next matrix
//          software-pipeline-prefetched and loaded transposed straight into
//          the A-layout; C->B repack via shfl).
// ---------------------------------------------------------------------------

#define T_TOTAL    2097152
#define NCHUNK     65536
#define CHUNK_L    32
#define NTILES     4             // CHUNK_L / 8
#define MAT_STRIDE 72            // 64 matrix + 1 logscale + pad (16B aligned)
#define WS_TL      0             // Tlin row-major [64]
#define WS_PI      64            // pi_lin [8]
#define WS_A       72            // emission a[8]
#define WS_B       80            // emission b[8][4]
#define WS_C       112           // emission c[8][4]
#define WS_CHUNK   160
#define WS_L1      (WS_CHUNK + NCHUNK * MAT_STRIDE)
#define WS_L2      (WS_L1 + 1024 * MAT_STRIDE)

#define LOG_2PI_F  1.8378770664093453f
#define LN2_F      0.69314718055994530942f

typedef float v2f __attribute__((ext_vector_type(2)));
typedef float v8f __attribute__((ext_vector_type(8)));

__device__ __forceinline__ float uni_f(float v) {
  return __uint_as_float(__builtin_amdgcn_readfirstlane(__float_as_uint(v)));
}

// Exact power-of-two rescale helpers: m = f * 2^e with f in [0.5, 1).
__device__ __forceinline__ int exp_of(float m) {
  return (int)((__float_as_uint(m) >> 23) & 0xffu) - 126;
}
__device__ __forceinline__ float pow2_neg(int e) {   // returns 2^(-e) exactly
  return __uint_as_float((unsigned)(127 - e) << 23);
}

// ---------------------------------------------------------------------------
// Setup: row log-softmax of trans -> Tlin, pi_lin, emission quadratic coeffs.
// lem_k(x) = a_k + sum_d (b_kd + c_kd*x_d)*x_d
// ---------------------------------------------------------------------------
__global__ void hmm_setup_kernel(const float* __restrict__ utrans,
                                 const float* __restrict__ upi,
                                 const float* __restrict__ means,
                                 const float* __restrict__ log_std,
                                 float* __restrict__ ws) {
  if (threadIdx.x != 0 || blockIdx.x != 0) return;
  for (int i = 0; i < 8; ++i) {
    float mx = utrans[i * 8];
    for (int j = 1; j < 8; ++j) mx = fmaxf(mx, utrans[i * 8 + j]);
    float s = 0.f;
    for (int j = 0; j < 8; ++j) s += __expf(utrans[i * 8 + j] - mx);
    float lse = mx + __logf(s);
    for (int j = 0; j < 8; ++j) ws[WS_TL + i * 8 + j] = __expf(utrans[i * 8 + j] - lse);
  }
  {
    float mx = upi[0];
    for (int j = 1; j < 8; ++j) mx = fmaxf(mx, upi[j]);
    float s = 0.f;
    for (int j = 0; j < 8; ++j) s += __expf(upi[j] - mx);
    float lse = mx + __logf(s);
    for (int j = 0; j < 8; ++j) ws[WS_PI + j] = __expf(upi[j] - lse);
  }
  for (int k = 0; k < 8; ++k) {
    float a = 0.f;
    for (int d = 0; d < 4; ++d) {
      float ls = log_std[k * 4 + d];
      float m  = means[k * 4 + d];
      float i2 = __expf(-2.f * ls);         // 1/std^2
      ws[WS_C + k * 4 + d] = -0.5f * i2;
      ws[WS_B + k * 4 + d] = m * i2;
      a += -0.5f * LOG_2PI_F - ls - 0.5f * m * m * i2;
    }
    ws[WS_A + k] = a;
  }
}

// ---------------------------------------------------------------------------
// Main scan: 2048 waves, 32 chunks per wave (one per lane), 32 steps/chunk.
// Each wave stages its 1024 contiguous timesteps (16KB) through LDS in
// double-buffered 4KB tiles using async global->LDS copies.
// ---------------------------------------------------------------------------
__global__ void __launch_bounds__(256)
hmm_scan_kernel(const float4* __restrict__ x4, float* __restrict__ ws) {
  __shared__ float4 smem[4096];             // 8 waves * 2 bufs * 32 chunks * 8 steps
  const int lane = threadIdx.x & 31;
  const int winb = threadIdx.x >> 5;
  const int wid  = blockIdx.x * 8 + winb;   // 0..2047
  const unsigned wavebase = (unsigned)(uintptr_t)(&smem[0]) + (unsigned)winb * 512u * 16u;

  // Transition matrix, forced uniform -> SGPR file (v_fmac with SGPR src0).
  float Tl[64];
#pragma unroll
  for (int i = 0; i < 64; ++i) Tl[i] = uni_f(ws[WS_TL + i]);
  // Emission coefficients (broadcast VGPRs).
  float ea[8], eb[32], ec[32];
#pragma unroll
  for (int i = 0; i < 8; ++i)  ea[i] = ws[WS_A + i];
#pragma unroll
  for (int i = 0; i < 32; ++i) eb[i] = ws[WS_B + i];
#pragma unroll
  for (int i = 0; i < 32; ++i) ec[i] = ws[WS_C + i];

  float P[8][8];
#pragma unroll
  for (int i = 0; i < 8; ++i)
#pragma unroll
    for (int j = 0; j < 8; ++j) P[i][j] = (i == j) ? 1.f : 0.f;
  float lsc = 0.f;          // natural-log scale from emission shifts
  int   esc = 0;            // exact base-2 exponent from renormalization

  const size_t W0 = (size_t)wid * (32 * CHUNK_L);  // wave's first timestep
  const int lm = lane & 7;                  // element within chunk-tile piece
  const int cb = lane >> 3;                 // chunk sub-index 0..3

  // Prefetch tile 0 (8 async b128 instrs; instr m covers chunks 4m..4m+3).
#pragma unroll
  for (int m = 0; m < 8; ++m) {
    int c = m * 4 + cb;
    const float4* g = x4 + (W0 + (size_t)c * CHUNK_L + (size_t)lm);
    unsigned ldst = wavebase + (unsigned)(c * 8 + lm) * 16u;
    asm volatile("global_load_async_to_lds_b128 %0, %1, off" :: "v"(ldst), "v"(g) : "memory");
  }

  const bool firstlane = (wid == 0) && (lane == 0);

  for (int k = 0; k < NTILES; ++k) {        // tiles of 8 steps
    if (k < NTILES - 1) {
      unsigned nboff = (unsigned)(((k + 1) & 1) * 256 * 16);
#pragma unroll
      for (int m = 0; m < 8; ++m) {
        int c = m * 4 + cb;
        const float4* g = x4 + (W0 + (size_t)c * CHUNK_L + (size_t)((k + 1) * 8 + lm));
        unsigned ldst = wavebase + nboff + (unsigned)(c * 8 + lm) * 16u;
        asm volatile("global_load_async_to_lds_b128 %0, %1, off" :: "v"(ldst), "v"(g) : "memory");
      }
      asm volatile("s_wait_asynccnt 0x8" ::: "memory");   // tile k complete
    } else {
      asm volatile("s_wait_asynccnt 0x0" ::: "memory");
    }

    const float4* buf = smem + winb * 512 + (k & 1) * 256 + lane * 8;
    for (int j = 0; j < 8; ++j) {
      float4 xv = buf[j];
      float lem[8];
#pragma unroll
      for (int q = 0; q < 8; ++q) {
        float a = ea[q];
        float t;
        t = fmaf(ec[q * 4 + 0], xv.x, eb[q * 4 + 0]); a = fmaf(t, xv.x, a);
        t = fmaf(ec[q * 4 + 1], xv.y, eb[q * 4 + 1]); a = fmaf(t, xv.y, a);
        t = fmaf(ec[q * 4 + 2], xv.z, eb[q * 4 + 2]); a = fmaf(t, xv.z, a);
        t = fmaf(ec[q * 4 + 3], xv.w, eb[q * 4 + 3]); a = fmaf(t, xv.w, a);
        lem[q] = a;
      }
      float mx = lem[0];
#pragma unroll
      for (int q = 1; q < 8; ++q) mx = fmaxf(mx, lem[q]);
      float em[8];
#pragma unroll
      for (int q = 0; q < 8; ++q) em[q] = __expf(lem[q] - mx);
      lsc += mx;

      if (firstlane && k == 0 && j == 0) {
        // t==0: M_0 = diag(em_0) (pi folded at the very end)
#pragma unroll
        for (int i = 0; i < 8; ++i)
#pragma unroll
          for (int jj = 0; jj < 8; ++jj) P[i][jj] *= em[jj];
      } else {
        // P <- (P @ Tlin) * diag(em), row-wise (in-place per row)
#pragma unroll
        for (int i = 0; i < 8; ++i) {
          float r[8];
#pragma unroll
          for (int jj = 0; jj < 8; ++jj) r[jj] = P[i][0] * Tl[jj];
#pragma unroll
          for (int kk = 1; kk < 8; ++kk)
#pragma unroll
            for (int jj = 0; jj < 8; ++jj) r[jj] = fmaf(P[i][kk], Tl[kk * 8 + jj], r[jj]);
#pragma unroll
          for (int jj = 0; jj < 8; ++jj) P[i][jj] = r[jj] * em[jj];
        }
      }
    }

    // Exact power-of-two renormalization every 8 steps (entries >= 0, max > 0).
    float m = P[0][0];
#pragma unroll
    for (int i = 0; i < 8; ++i)
#pragma unroll
      for (int j = 0; j < 8; ++j) m = fmaxf(m, P[i][j]);
    int e = exp_of(m);
    float inv = pow2_neg(e);
#pragma unroll
    for (int i = 0; i < 8; ++i)
#pragma unroll
      for (int j = 0; j < 8; ++j) P[i][j] *= inv;
    esc += e;
  }

  float* o = ws + WS_CHUNK + (size_t)(wid * 32 + lane) * MAT_STRIDE;
#pragma unroll
  for (int i = 0; i < 8; ++i) {
    *(float4*)(o + i * 8)     = make_float4(P[i][0], P[i][1], P[i][2], P[i][3]);
    *(float4*)(o + i * 8 + 4) = make_float4(P[i][4], P[i][5], P[i][6], P[i][7]);
  }
  o[64] = lsc + (float)esc * LN2_F;
}

// ---------------------------------------------------------------------------
// WMMA reduction: each wave combines R consecutive matrices in chunk order.
// Carry Q = P^T in C-layout (8x8 block at M,N in [0,8)). Each step:
//   Q_new = Q_new_chunk @ Q  ->  A = P_c^T (loaded transposed), B = Q.
// Two chained v_wmma_f32_16x16x4_f32 cover K = 0..3 and 4..7.
// Next matrix is prefetched one iteration ahead (clamped index, branch-free)
// so the serial product chain is not load-latency bound.
// Live data sits in lanes 0..7 only -> shuffle reductions use xor 4,2,1.
// ---------------------------------------------------------------------------
__global__ void hmm_reduce_kernel(const float* __restrict__ in, float* __restrict__ out,
                                  int n, int R, int finalize,
                                  const float* __restrict__ pi, float* __restrict__ dout) {
  const int wid  = (int)((blockIdx.x * blockDim.x + threadIdx.x) >> 5);
  const int lane = threadIdx.x & 31;
  const int start = wid * R;
  if (start >= n) return;
  const int cnt = min(R, n - start);

  const bool hi = lane >= 16;
  const bool valid = (lane & 15) < 8;
  const int mc = lane & 7;
  const int r0 = hi ? 2 : 0;   // A1 holds P rows {0,1} lo-half / {2,3} hi-half
  const int r4 = hi ? 6 : 4;   // A2 holds P rows {4,5} lo-half / {6,7} hi-half

  // Init Q from first matrix: acc[r] at lane n<8 = Q[r][n] = P[n][r].
  const float4* m04 = (const float4*)(in + (size_t)start * MAT_STRIDE);
  float4 qa = m04[mc * 2];
  float4 qb = m04[mc * 2 + 1];
  v8f acc;
  acc[0] = (lane < 8) ? qa.x : 0.f;  acc[1] = (lane < 8) ? qa.y : 0.f;
  acc[2] = (lane < 8) ? qa.z : 0.f;  acc[3] = (lane < 8) ? qa.w : 0.f;
  acc[4] = (lane < 8) ? qb.x : 0.f;  acc[5] = (lane < 8) ? qb.y : 0.f;
  acc[6] = (lane < 8) ? qb.z : 0.f;  acc[7] = (lane < 8) ? qb.w : 0.f;
  float ls = in[(size_t)start * MAT_STRIDE + 64];
  int   esc = 0;

  // Prefetch matrix start+1.
  float t0 = 0.f, t1 = 0.f, t2 = 0.f, t3 = 0.f, lsn = 0.f;
  if (cnt > 1) {
    const float* mi = in + (size_t)(start + 1) * MAT_STRIDE;
    t0 = mi[r0 * 8 + mc];
    t1 = mi[(r0 + 1) * 8 + mc];
    t2 = mi[r4 * 8 + mc];
    t3 = mi[(r4 + 1) * 8 + mc];
    lsn = mi[64];
  }

  for (int i = 1; i < cnt; ++i) {
    // Current iteration's prefetched operand.
    float u0 = t0, u1 = t1, u2 = t2, u3 = t3;
    ls += lsn;
    // Issue next iteration's loads (clamped; overlaps with WMMA below).
    {
      int nxt = i + 1 < cnt ? i + 1 : cnt - 1;
      const float* mi = in + (size_t)(start + nxt) * MAT_STRIDE;
      t0 = mi[r0 * 8 + mc];
      t1 = mi[(r0 + 1) * 8 + mc];
      t2 = mi[r4 * 8 + mc];
      t3 = mi[(r4 + 1) * 8 + mc];
      lsn = mi[64];
    }

    // A = P_c^T in A-layout (rows m>=8 zeroed).
    v2f a1, a2;
    a1[0] = valid ? u0 : 0.f;  a1[1] = valid ? u1 : 0.f;
    a2[0] = valid ? u2 : 0.f;  a2[1] = valid ? u3 : 0.f;

    // B = Q from C-layout acc: B1 = {K0|K2, K1|K3}, B2 = {K4|K6, K5|K7}.
    float s2 = __shfl_xor(acc[2], 16, 32);
    float s3 = __shfl_xor(acc[3], 16, 32);
    float s6 = __shfl_xor(acc[6], 16, 32);
    float s7 = __shfl_xor(acc[7], 16, 32);
    v2f b1, b2;
    b1[0] = hi ? s2 : acc[0];  b1[1] = hi ? s3 : acc[1];
    b2[0] = hi ? s6 : acc[4];  b2[1] = hi ? s7 : acc[5];

    v8f c = {0.f, 0.f, 0.f, 0.f, 0.f, 0.f, 0.f, 0.f};
    c = __builtin_amdgcn_wmma_f32_16x16x4_f32(false, a1, false, b1, (short)0, c, false, false);
    c = __builtin_amdgcn_wmma_f32_16x16x4_f32(false, a2, false, b2, (short)0, c, false, false);
    acc = c;

    // Exact power-of-two renorm; data only in lanes 0..7 -> xor 4,2,1.
    float mx = acc[0];
#pragma unroll
    for (int r = 1; r < 8; ++r) mx = fmaxf(mx, acc[r]);
    mx = fmaxf(mx, __shfl_xor(mx, 4, 32));
    mx = fmaxf(mx, __shfl_xor(mx, 2, 32));
    mx = fmaxf(mx, __shfl_xor(mx, 1, 32));
    int e = exp_of(mx);
    float inv = pow2_neg(e);
#pragma unroll
    for (int r = 0; r < 8; ++r) acc[r] *= inv;
    esc += e;
  }

  float lstot = ls + (float)esc * LN2_F;
  if (!finalize) {
    // Store as P row-major: lane i<8 holds row i of P across acc[0..7].
    float* o = out + (size_t)wid * MAT_STRIDE;
    if (lane < 8) {
      *(float4*)(o + lane * 8)     = make_float4(acc[0], acc[1], acc[2], acc[3]);
      *(float4*)(o + lane * 8 + 4) = make_float4(acc[4], acc[5], acc[6], acc[7]);
    }
    if (lane == 0) o[64] = lstot;
  } else {
    // ans = log( sum_{r,n} Q[r][n] * pi[n] ) + total logscale
    float p = (lane < 8) ? pi[mc] : 0.f;
    float part = 0.f;
#pragma unroll
    for (int r = 0; r < 8; ++r) part = fmaf(acc[r], p, part);
    part += __shfl_xor(part, 4, 32);
    part += __shfl_xor(part, 2, 32);
    part += __shfl_xor(part, 1, 32);
    if (lane == 0) dout[0] = __logf(part) + lstot;
  }
}

// ---------------------------------------------------------------------------
extern "C" void kernel_launch(void* const* d_in, const int* in_sizes, int n_in,
                              void* d_out, int out_size, void* d_ws, size_t ws_size,
                              hipStream_t stream) {
  const float* x_seq  = (const float*)d_in[0];   // [T,4] f32
  const float* utrans = (const float*)d_in[1];   // [8,8]
  const float* upi    = (const float*)d_in[2];   // [8]
  const float* means  = (const float*)d_in[3];   // [8,4]
  const float* lstd   = (const float*)d_in[4];   // [8,4]
  float* ws  = (float*)d_ws;
  float* out = (float*)d_out;

  hmm_setup_kernel<<<1, 32, 0, stream>>>(utrans, upi, means, lstd, ws);
  hmm_scan_kernel<<<256, 256, 0, stream>>>((const float4*)x_seq, ws);
  // 65536 -> 1024 -> 16 -> 1 (in chunk order, 64/64/16-way per wave)
  hmm_reduce_kernel<<<128, 256, 0, stream>>>(ws + WS_CHUNK, ws + WS_L1, NCHUNK, 64, 0, nullptr, nullptr);
  hmm_reduce_kernel<<<2, 256, 0, stream>>>(ws + WS_L1, ws + WS_L2, 1024, 64, 0, nullptr, nullptr);
  hmm_reduce_kernel<<<1, 32, 0, stream>>>(ws + WS_L2, nullptr, 16, 16, 1, ws + WS_PI, out);
}